// clDice_14620068676019
// MI455X (gfx1250) — compile-verified
//
#include <hip/hip_runtime.h>

// ---------------------------------------------------------------------------
// soft-clDice loss for (8,1,512,512) f32 inputs on MI455X (gfx1250).
// Strategy: ping-pong the 16 image planes + skel (48 MB total, L2-resident on
// a 192MB-L2 part) through one fused stencil kernel per soft_skel iteration.
// Each workgroup pulls its halo'd 38x38 tile into LDS with a single TDM
// tensor_load_to_lds, computes erode / erode / dilate stages in LDS with
// workgroup barriers, and writes back the 32x32 interior (img' and skel')
// with two TDM tensor_store_from_lds ops from packed LDS staging tiles.
// Final reductions are deterministic fixed-order LDS trees (no atomics).
// ---------------------------------------------------------------------------

constexpr int IMG_H   = 512;
constexpr int IMG_W   = 512;
constexpr int PLANE   = IMG_H * IMG_W;   // 262144
constexpr int NPLANES = 16;              // planes 0..7 = sigmoid(output), 8..15 = target
constexpr int NSAMP   = 8;
constexpr int TILE    = 32;

// LDS layout (floats) inside the single static array (LDS offset 0):
constexpr int OFF_TILE = 0;      // up to 38*38 = 1444, packed rows of stride w
constexpr int OFF_E    = 1444;   // 36 rows, stride 37 (1332)
constexpr int OFF_E2   = 2776;   // 34 rows, stride 35 (1190)
constexpr int OFF_PE   = 3968;   // packed 32x32 E tile for TDM store
constexpr int OFF_PS   = 4992;   // packed 32x32 skel tile for TDM store
constexpr int LDS_FLTS = 6016;   // 24,064 bytes

typedef __attribute__((ext_vector_type(4))) unsigned int v4u;
typedef __attribute__((ext_vector_type(8))) int          v8i;
typedef __attribute__((ext_vector_type(4))) int          v4i;

__device__ __forceinline__ float sigm(float x) { return 1.0f / (1.0f + __expf(-x)); }

#if __has_builtin(__builtin_amdgcn_tensor_load_to_lds)
// Build D# group0: count=1 | lds_byte_addr | global_addr[56:0] | type=2
__device__ __forceinline__ v4u tdm_g0(unsigned lds_byte, const void* gptr)
{
    unsigned long long ga = (unsigned long long)(size_t)gptr;
    v4u g0 = { 1u, lds_byte,
               (unsigned)(ga & 0xFFFFFFFFu),
               (unsigned)((ga >> 32) & 0x01FFFFFFu) | 0x80000000u };
    return g0;
}
// Build D# group1 for a 2-D f32 tile: tensor 512x512, row stride 512 elements
__device__ __forceinline__ v8i tdm_g1(int tile_w, int tile_h)
{
    v8i g1 = { (int)(2u << 16),            // data_size = 4 bytes
               (int)(512u << 16),          // tensor_dim0 = 512 (bits 79:48, low half)
               (int)(512u << 16),          // tensor_dim1 = 512 (bits 111:80, low half)
               (int)((unsigned)tile_w << 16), // tile_dim0
               tile_h,                     // tile_dim1 (tile_dim2 = 0)
               IMG_W,                      // tensor_dim0_stride = 512
               0, 0 };
    return g1;
}
#endif

// ---------------------------------------------------------------------------
// One soft_skel step, fused:
//   E  = apply_erode ? erode(src) : src          (plus-stencil min, +inf pad)
//   E2 = erode(E);  D = dilate3x3(E2)            (-inf pad)
//   delta = relu(E - D)
//   skel  = init_skel ? delta : skel + relu(delta - skel*delta)
//   dst   = E
// Tile: 32x32 interior, halo 3.  TDM loads the (clamped) halo tile into LDS
// and TDM-stores the packed interior results.
// ---------------------------------------------------------------------------
__global__ void __launch_bounds__(256)
skel_iter_k(const float* __restrict__ src, float* __restrict__ dst,
            float* __restrict__ skel, int apply_erode, int init_skel)
{
    const float INF = __builtin_inff();
    __shared__ float smem[LDS_FLTS];
    float* tilebuf = smem + OFF_TILE;
    float* eE      = smem + OFF_E;
    float* eE2     = smem + OFF_E2;
    float* packE   = smem + OFF_PE;
    float* packS   = smem + OFF_PS;

    const int tid = threadIdx.x;
    const int X0  = blockIdx.x * TILE;
    const int Y0  = blockIdx.y * TILE;
    const int pl  = blockIdx.z;
    const float* sp = src + pl * PLANE;
    float* dp = dst  + pl * PLANE;
    float* kp = skel + pl * PLANE;

    // halo extent clamped to the image: every in-image tap lands in [ys,ye)x[xs,xe)
    const int ys = max(Y0 - 3, 0), ye = min(Y0 + TILE + 3, IMG_H);
    const int xs = max(X0 - 3, 0), xe = min(X0 + TILE + 3, IMG_W);
    const int h = ye - ys, w = xe - xs;

#if __has_builtin(__builtin_amdgcn_tensor_load_to_lds)
    if (tid < 32) {
        v4u g0 = tdm_g0(OFF_TILE * 4u, sp + ys * IMG_W + xs);
        v8i g1 = tdm_g1(w, h);
        v4i gz4 = { 0, 0, 0, 0 };           // 2-D tensor: groups 2/3 unused
        v8i gz8 = { 0, 0, 0, 0, 0, 0, 0, 0 };
        __builtin_amdgcn_tensor_load_to_lds(g0, g1, gz4, gz4, gz8, 0);
    }
#else
    for (int i = tid; i < h * w; i += 256) {
        int r = i / w;
        tilebuf[i] = sp[(ys + r) * IMG_W + xs + (i - r * w)];
    }
#endif

    // overlap TDM with the skel register preload (interior only, coalesced)
    const int lx  = tid & 31;
    const int lyb = tid >> 5;   // 0..7
    float skv[4] = {0.f, 0.f, 0.f, 0.f};
    if (!init_skel) {
        #pragma unroll
        for (int r = 0; r < 4; ++r)
            skv[r] = kp[(Y0 + lyb + 8 * r) * IMG_W + X0 + lx];
    }

#if __has_builtin(__builtin_amdgcn_s_wait_tensorcnt)
    if (tid < 32) __builtin_amdgcn_s_wait_tensorcnt(0);
#endif
    __syncthreads();

    // ---- stage E: extent [Y0-2, Y0+34) x [X0-2, X0+34), OOB-of-image -> +inf
    for (int i = tid; i < 36 * 36; i += 256) {
        int ly = i / 36, lc = i - ly * 36;
        int gy = Y0 - 2 + ly, gx = X0 - 2 + lc;
        float e;
        if (gy >= 0 && gy < IMG_H && gx >= 0 && gx < IMG_W) {
            float c = tilebuf[(gy - ys) * w + (gx - xs)];
            if (apply_erode) {
                float up = (gy - 1 >= 0)    ? tilebuf[(gy - 1 - ys) * w + (gx - xs)] : INF;
                float dn = (gy + 1 < IMG_H) ? tilebuf[(gy + 1 - ys) * w + (gx - xs)] : INF;
                float lf = (gx - 1 >= 0)    ? tilebuf[(gy - ys) * w + (gx - 1 - xs)] : INF;
                float rt = (gx + 1 < IMG_W) ? tilebuf[(gy - ys) * w + (gx + 1 - xs)] : INF;
                e = fminf(fminf(fminf(up, dn), fminf(lf, rt)), c);
            } else {
                e = c;
            }
        } else {
            e = INF;                 // identity for the next erode
        }
        eE[ly * 37 + lc] = e;
    }
    __syncthreads();

    // ---- stage E2 = erode(E): extent [Y0-1, Y0+33)^2, OOB-of-image -> -inf
    for (int i = tid; i < 34 * 34; i += 256) {
        int ly = i / 34, lc = i - ly * 34;
        int gy = Y0 - 1 + ly, gx = X0 - 1 + lc;
        float e2;
        if (gy >= 0 && gy < IMG_H && gx >= 0 && gx < IMG_W) {
            int ey = ly + 1, ex = lc + 1;      // offset into eE extent
            float c  = eE[ey * 37 + ex];
            float up = eE[(ey - 1) * 37 + ex];
            float dn = eE[(ey + 1) * 37 + ex];
            float lf = eE[ey * 37 + ex - 1];
            float rt = eE[ey * 37 + ex + 1];
            e2 = fminf(fminf(fminf(up, dn), fminf(lf, rt)), c);
        } else {
            e2 = -INF;               // identity for the dilate
        }
        eE2[ly * 35 + lc] = e2;
    }
    __syncthreads();

    // ---- interior: D = dilate3x3(E2), delta, skel update -> packed LDS tiles
    #pragma unroll
    for (int r = 0; r < 4; ++r) {
        int y  = lyb + 8 * r;
        float E = eE[(y + 2) * 37 + (lx + 2)];
        int ey = y + 1, ex = lx + 1;
        float D = -INF;
        #pragma unroll
        for (int dy = -1; dy <= 1; ++dy)
            #pragma unroll
            for (int dx = -1; dx <= 1; ++dx)
                D = fmaxf(D, eE2[(ey + dy) * 35 + (ex + dx)]);
        float delta = fmaxf(E - D, 0.0f);
        float sk = init_skel ? delta
                             : (skv[r] + fmaxf(delta - skv[r] * delta, 0.0f));
        packE[y * TILE + lx] = E;
        packS[y * TILE + lx] = sk;
    }
    __syncthreads();

    // ---- write-back: two 2-D TDM stores of the packed 32x32 tiles.
    // S_ENDPGM's implicit wait-idle covers TENSORcnt; the kernel boundary
    // orders these against the next iteration's TDM loads.
#if __has_builtin(__builtin_amdgcn_tensor_store_from_lds)
    if (tid < 32) {
        v4i gz4 = { 0, 0, 0, 0 };
        v8i gz8 = { 0, 0, 0, 0, 0, 0, 0, 0 };
        v8i g1  = tdm_g1(TILE, TILE);
        v4u g0e = tdm_g0(OFF_PE * 4u, dp + Y0 * IMG_W + X0);
        __builtin_amdgcn_tensor_store_from_lds(g0e, g1, gz4, gz4, gz8, 0);
        v4u g0s = tdm_g0(OFF_PS * 4u, kp + Y0 * IMG_W + X0);
        __builtin_amdgcn_tensor_store_from_lds(g0s, g1, gz4, gz4, gz8, 0);
    }
#else
    for (int i = tid; i < TILE * TILE; i += 256) {
        int gidx = (Y0 + (i >> 5)) * IMG_W + X0 + (i & 31);
        dp[gidx] = packE[i];
        kp[gidx] = packS[i];
    }
#endif
}

// ---------------------------------------------------------------------------
// imgA[plane] = sigmoid(output) for planes 0..7, target for planes 8..15
// ---------------------------------------------------------------------------
__global__ void init_img_k(const float* __restrict__ outp,
                           const float* __restrict__ tgt,
                           float* __restrict__ img)
{
    int idx   = blockIdx.x * 256 + threadIdx.x;
    int plane = idx / PLANE;
    int i     = idx - plane * PLANE;
    img[idx] = (plane < NSAMP) ? sigm(outp[plane * PLANE + i])
                               : tgt[(plane - NSAMP) * PLANE + i];
}

// ---------------------------------------------------------------------------
// Stage-1 reduction: grid (64 chunks, 8 samples), each block reduces 4096
// contiguous elements of its sample into 7 partial sums (fixed-tree, no atomics)
// ---------------------------------------------------------------------------
__global__ void reduce1_k(const float* __restrict__ outp,
                          const float* __restrict__ tgt,
                          const float* __restrict__ skel,
                          float* __restrict__ partials)
{
    __shared__ float red[256];
    const int s = blockIdx.y, b = blockIdx.x, tid = threadIdx.x;
    const int base = s * PLANE + b * 4096;
    const float* sp_ = skel + s * PLANE + b * 4096;            // s_p
    const float* sl_ = skel + (s + NSAMP) * PLANE + b * 4096;  // s_l

    float a1 = 0, a2 = 0, b1 = 0, b2 = 0, in_ = 0, ps = 0, ts = 0;
    for (int j = tid; j < 4096; j += 256) {
        float p = sigm(outp[base + j]);
        float t = tgt[base + j];
        float spv = sp_[j], slv = sl_[j];
        a1 += spv * t;  a2 += spv;
        b1 += slv * p;  b2 += slv;
        in_ += p * t;   ps += p;   ts += t;
    }
    float vals[7] = { a1, a2, b1, b2, in_, ps, ts };
    for (int k = 0; k < 7; ++k) {
        red[tid] = vals[k];
        __syncthreads();
        for (int off = 128; off > 0; off >>= 1) {
            if (tid < off) red[tid] += red[tid + off];
            __syncthreads();
        }
        if (tid == 0) partials[(s * 64 + b) * 8 + k] = red[0];
        __syncthreads();
    }
}

__global__ void reduce2_k(const float* __restrict__ partials,
                          float* __restrict__ sums)
{
    __shared__ float red[64];
    const int s = blockIdx.x, t = threadIdx.x;
    for (int k = 0; k < 7; ++k) {
        red[t] = partials[(s * 64 + t) * 8 + k];
        __syncthreads();
        for (int off = 32; off > 0; off >>= 1) {
            if (t < off) red[t] += red[t + off];
            __syncthreads();
        }
        if (t == 0) sums[s * 8 + k] = red[0];
        __syncthreads();
    }
}

__global__ void final_k(const float* __restrict__ sums, float* __restrict__ outv)
{
    float cl = 0.f, di = 0.f;
    for (int s = 0; s < NSAMP; ++s) {
        float a1 = sums[s * 8 + 0], a2 = sums[s * 8 + 1];
        float b1 = sums[s * 8 + 2], b2 = sums[s * 8 + 3];
        float I  = sums[s * 8 + 4], P  = sums[s * 8 + 5], T = sums[s * 8 + 6];
        float tprec = (a1 + 1.0f) / (a2 + 1.0f);
        float tsens = (b1 + 1.0f) / (b2 + 1.0f);
        cl += 1.0f - 2.0f * tprec * tsens / (tprec + tsens);
        di += 1.0f - 2.0f * (I + 1.0f) / (P + T + 1.0f);
    }
    outv[0] = 0.7f * (di / 8.0f) + 0.3f * (cl / 8.0f);
}

// ---------------------------------------------------------------------------
extern "C" void kernel_launch(void* const* d_in, const int* in_sizes, int n_in,
                              void* d_out, int out_size, void* d_ws, size_t ws_size,
                              hipStream_t stream)
{
    (void)in_sizes; (void)n_in; (void)out_size; (void)ws_size;
    const float* outp = (const float*)d_in[0];
    const float* tgt  = (const float*)d_in[1];

    float* imgA     = (float*)d_ws;                 // 16 MB
    float* imgB     = imgA + NPLANES * PLANE;       // 16 MB
    float* skel     = imgB + NPLANES * PLANE;       // 16 MB
    float* partials = skel + NPLANES * PLANE;       // 8*64*8 floats
    float* sums     = partials + 8 * 64 * 8;        // 8*8 floats

    init_img_k<<<NPLANES * PLANE / 256, 256, 0, stream>>>(outp, tgt, imgA);

    dim3 g(IMG_W / TILE, IMG_H / TILE, NPLANES);
    // init step: skel = relu(img - open(img)), dst = img copy
    skel_iter_k<<<g, 256, 0, stream>>>(imgA, imgB, skel, /*erode=*/0, /*init=*/1);
    float* src = imgB; float* dst = imgA;
    for (int it = 0; it < 50; ++it) {
        skel_iter_k<<<g, 256, 0, stream>>>(src, dst, skel, /*erode=*/1, /*init=*/0);
        float* t2 = src; src = dst; dst = t2;
    }

    reduce1_k<<<dim3(64, NSAMP), 256, 0, stream>>>(outp, tgt, skel, partials);
    reduce2_k<<<NSAMP, 64, 0, stream>>>(partials, sums);
    final_k<<<1, 1, 0, stream>>>(sums, (float*)d_out);
}